// TAE2d_5514738008774
// MI455X (gfx1250) — compile-verified
//
#include <hip/hip_runtime.h>
#include <math.h>

#define N_HEAD 16
#define D_K    8
#define D_IN   256
#define SZ_B   16384
#define SEQ    45
#define SEQ_PAD 48
#define VSTRIDE 260   // LDS row stride (floats): 260 % 64 = 4 -> conflict-free WMMA A loads

typedef __attribute__((ext_vector_type(2))) float v2f;
typedef __attribute__((ext_vector_type(4))) float v4f;
typedef __attribute__((ext_vector_type(8))) float v8f;

__device__ __forceinline__ unsigned lds_addr_of(const void* p) {
    // low 32 bits of a flat shared-aperture address are the LDS byte address
    return (unsigned)(uintptr_t)p;
}

__device__ __forceinline__ void async_load_b128_to_lds(unsigned lds_byte, const void* gaddr) {
    asm volatile("global_load_async_to_lds_b128 %0, %1, off"
                 :: "v"(lds_byte), "v"(gaddr)
                 : "memory");
}

// ---- prep: fold shared query into key projection ----------------------------
// Wq[h][i] = (1/sqrt(d_k)) * sum_d Q[h][d] * Wk[h*8+d][i]   (pair-interleaved)
// bq[h]    = (1/sqrt(d_k)) * sum_d Q[h][d] * bk[h*8+d]
__global__ void tae_prep(const float* __restrict__ Q,
                         const float* __restrict__ Wk,
                         const float* __restrict__ bk,
                         float* __restrict__ wq,   // [16*256] pair-interleaved
                         float* __restrict__ bq)   // [16]
{
    const float inv_temp = 0.35355339059327373f; // 1/sqrt(8)
    const int h = blockIdx.x;
    const int i = threadIdx.x;
    float acc = 0.f;
    #pragma unroll
    for (int d = 0; d < D_K; ++d)
        acc += Q[h*D_K + d] * Wk[(h*D_K + d)*D_IN + i];
    // element (h, i) stored at dword (i/2)*32 + h*2 + (i&1)  -> b64-loadable pairs
    wq[(i >> 1)*32 + h*2 + (i & 1)] = acc * inv_temp;
    if (i == 0) {
        float bb = 0.f;
        #pragma unroll
        for (int d = 0; d < D_K; ++d) bb += Q[h*D_K + d] * bk[h*D_K + d];
        bq[h] = bb * inv_temp;
    }
}

// ---- main: one workgroup per batch element ----------------------------------
__global__ __launch_bounds__(256)
void tae_main(const float* __restrict__ v,
              const float* __restrict__ wq,
              const float* __restrict__ bq,
              float* __restrict__ out,     // [B, 256]
              float* __restrict__ attn)    // [16, B, 45]
{
    __shared__ __attribute__((aligned(16))) float v_lds[SEQ_PAD * VSTRIDE];
    __shared__ __attribute__((aligned(16))) float wq_lds[N_HEAD * D_IN];
    __shared__ float l_lds[SEQ_PAD * N_HEAD];    // logits -> attn

    const int b    = blockIdx.x;
    const int tid  = threadIdx.x;
    const int lane = tid & 31;
    const int w    = tid >> 5;

    // ---- stage v[b] (45x256 f32) into LDS via async DMA (no VGPR round-trip)
    {
        const float* vbase = v + (size_t)b * (SEQ * D_IN);
        for (int i = tid; i < SEQ * (D_IN/4); i += 256) {
            const int row = i >> 6, c4 = i & 63;
            async_load_b128_to_lds(lds_addr_of(&v_lds[row*VSTRIDE + c4*4]),
                                   vbase + i*4);
        }
        // zero-pad rows 45..47 (read by WMMA, masked out of softmax/agg)
        for (int i = tid; i < (SEQ_PAD - SEQ) * D_IN; i += 256)
            v_lds[(SEQ + (i >> 8))*VSTRIDE + (i & 255)] = 0.f;
        // stage Wq (16 KB, L2-resident) the same way
        for (int i = tid; i < (N_HEAD*D_IN)/4; i += 256)
            async_load_b128_to_lds(lds_addr_of(&wq_lds[i*4]), wq + i*4);
        asm volatile("s_wait_asynccnt 0x0" ::: "memory");
    }
    __syncthreads();

    // logits[48x16] = v @ WqT + bq  via V_WMMA_F32_16X16X4_F32
    // waves 0..2 each own one 16-row M-tile; bias folded into accumulator init
    if (w < 3) {
        const int mrow = 16*w + (lane & 15);
        const int koff = (lane >> 4) << 1;       // lanes 0-15: K,K+1 ; lanes 16-31: K+2,K+3
        const float bqv = bq[lane & 15];
        v8f c;
        #pragma unroll
        for (int r = 0; r < 8; ++r) c[r] = bqv;
        #pragma unroll 4
        for (int k0 = 0; k0 < D_IN; k0 += 4) {
            v2f a  = *(const v2f*)&v_lds[mrow*VSTRIDE + k0 + koff];
            v2f bb = *(const v2f*)&wq_lds[(k0 + koff)*16 + ((lane & 15) << 1)];
            c = __builtin_amdgcn_wmma_f32_16x16x4_f32(false, a, false, bb,
                                                      (short)0, c, false, false);
        }
        const int col   = lane & 15;
        const int rbase = 16*w + ((lane >> 4) << 3);   // C layout: lanes16-31 hold M=r+8
        #pragma unroll
        for (int r = 0; r < 8; ++r) l_lds[(rbase + r)*N_HEAD + col] = c[r];
    }
    __syncthreads();

    // softmax over time (s < 45 only; padded rows never read), one thread/head
    if (tid < N_HEAD) {
        float m = -1e30f;
        for (int s = 0; s < SEQ; ++s) m = fmaxf(m, l_lds[s*N_HEAD + tid]);
        float sum = 0.f;
        for (int s = 0; s < SEQ; ++s) {
            const float e = __expf(l_lds[s*N_HEAD + tid] - m);
            l_lds[s*N_HEAD + tid] = e;
            sum += e;
        }
        const float inv = 1.0f / sum;
        for (int s = 0; s < SEQ; ++s) l_lds[s*N_HEAD + tid] *= inv;
    }
    __syncthreads();

    // attn output: [16, B, 45], coalesced runs of 45 per head
    for (int i = tid; i < N_HEAD*SEQ; i += 256) {
        const int h = i / SEQ, s = i - h*SEQ;
        attn[(size_t)h * SZ_B * SEQ + (size_t)b * SEQ + s] = l_lds[s*N_HEAD + h];
    }

    // out[b, c] = sum_s attn[c>>4, s] * v[b, s, c]   (broadcast + stride-1 LDS)
    {
        const int c  = tid;
        const int h  = c >> 4;
        float acc = 0.f;
        for (int s = 0; s < SEQ; ++s)
            acc = fmaf(l_lds[s*N_HEAD + h], v_lds[s*VSTRIDE + c], acc);
        out[(size_t)b * D_IN + c] = acc;
    }
}

extern "C" void kernel_launch(void* const* d_in, const int* in_sizes, int n_in,
                              void* d_out, int out_size, void* d_ws, size_t ws_size,
                              hipStream_t stream) {
    const float* v  = (const float*)d_in[0];   // [16384, 45, 256]
    const float* Q  = (const float*)d_in[1];   // [16, 8]
    const float* Wk = (const float*)d_in[2];   // [128, 256]
    const float* bk = (const float*)d_in[3];   // [128]

    float* wq = (float*)d_ws;                  // 4096 floats
    float* bq = wq + N_HEAD * D_IN;            // 16 floats

    float* out  = (float*)d_out;                       // [B, 256]
    float* attn = out + (size_t)SZ_B * D_IN;           // [16, B, 45]

    tae_prep<<<N_HEAD, D_IN, 0, stream>>>(Q, Wk, bk, wq, bq);
    tae_main<<<SZ_B, 256, 0, stream>>>(v, wq, bq, out, attn);
}